// RGCNDGISapMetricLearning_24421184045371
// MI455X (gfx1250) — compile-verified
//
#include <hip/hip_runtime.h>
#include <hip/hip_bf16.h>

// Problem constants (from reference)
#define NN   8192
#define EE   262144
#define HH   768
#define RR   8
#define NBB  8
#define BB   2048
#define KREL (RR*HH)   // 6144

typedef __attribute__((ext_vector_type(2))) float v2f;
typedef __attribute__((ext_vector_type(8))) float v8f;

#if defined(__gfx1250__) && __has_builtin(__builtin_amdgcn_global_load_async_to_lds_b32) && __has_builtin(__builtin_amdgcn_s_wait_asynccnt)
#define HAVE_ASYNC_LDS 1
#else
#define HAVE_ASYNC_LDS 0
#endif

#define AS_GLOBAL __attribute__((address_space(1)))
#define AS_LOCAL  __attribute__((address_space(3)))

// ---------------------------------------------------------------------------
// w_rel[r,i,o] = sum_b comp[r,b] * basis[b,i,o]   -> stored as [R*H, H] row-major
__global__ void k_wrel(const float* __restrict__ comp, const float* __restrict__ basis,
                       float* __restrict__ wrel) {
    int idx = blockIdx.x * 256 + threadIdx.x;          // R*H*H = 4718592 exact
    int r   = idx / (HH * HH);
    int rem = idx % (HH * HH);
    float acc = 0.f;
#pragma unroll
    for (int b = 0; b < NBB; b++)
        acc += comp[r * NBB + b] * basis[(size_t)b * HH * HH + rem];
    wrel[idx] = acc;
}

// edge counts per (dst, rel) segment
__global__ void k_count(const int* __restrict__ dst, const int* __restrict__ rel,
                        unsigned* __restrict__ cnt) {
    int e = blockIdx.x * 256 + threadIdx.x;
    if (e < EE) atomicAdd(&cnt[dst[e] * RR + rel[e]], 1u);
}

// scatter-add messages: sums[(dst*R+rel)*H + h] += x[gather_row, h]
__global__ void k_scatter(const float* __restrict__ x, const int* __restrict__ src,
                          const int* __restrict__ dst, const int* __restrict__ rel,
                          const int* __restrict__ perm, float* __restrict__ sums) {
    int e = blockIdx.x;
    int s = src[e];
    if (perm) s = perm[s];
    size_t seg = (size_t)(dst[e] * RR + rel[e]) * HH;
    const float* xr = x + (size_t)s * HH;
    for (int h = threadIdx.x; h < HH; h += 256)
        atomicAdd(&sums[seg + h], xr[h]);
}

// sums -> mean in place
__global__ void k_mean(float* __restrict__ sums, const unsigned* __restrict__ cnt) {
    size_t idx = (size_t)blockIdx.x * 256 + threadIdx.x;   // N*R*H
    unsigned c = cnt[idx / HH];
    if (c == 0u) c = 1u;
    sums[idx] *= 1.0f / (float)c;
}

// ---------------------------------------------------------------------------
// Tiled f32 WMMA GEMM: C[M,768] (+)= A[M,K] @ B[K,768] (+ bias).
// Block = 256 thr (8 waves), tile = 128(M) x 64(N).
// B chunk staged TRANSPOSED in LDS: lbT[col*34 + k] (pad 34 -> 8B-aligned pairs,
// conflict-free ds_load_b64 feeding WMMA directly, no repack movs).
#define LB_STRIDE 34
__global__ __launch_bounds__(256) void k_gemm(
    const float* __restrict__ A, const float* __restrict__ Bm,
    float* __restrict__ C, const float* __restrict__ bias,
    const int* __restrict__ rowmap, int K, int acc_flag) {
    __shared__ float lbT[64 * LB_STRIDE];
    const int tid   = threadIdx.x;
    const int wave  = tid >> 5, lane = tid & 31;
    const int lrow  = lane & 15;
    const int khalf = (lane < 16) ? 0 : 2;
    const int mbase = blockIdx.x * 128 + wave * 16;
    const int nbase = blockIdx.y * 64;

    int arow = mbase + lrow;
    if (rowmap) arow = rowmap[arow];
    const float* Arow = A + (size_t)arow * K;

    const v8f vzero = {0.f,0.f,0.f,0.f,0.f,0.f,0.f,0.f};
    v8f acc[4] = {vzero, vzero, vzero, vzero};

    for (int kc = 0; kc < K; kc += 32) {
        // stage B chunk [kc..kc+31] x [nbase..nbase+63], transposed into LDS
#if HAVE_ASYNC_LDS
#pragma unroll
        for (int i = 0; i < 8; i++) {
            int e = tid + i * 256;
            int br = e >> 6, bc = e & 63;
            __builtin_amdgcn_global_load_async_to_lds_b32(
                (AS_GLOBAL int*)&Bm[(size_t)(kc + br) * HH + nbase + bc],
                (AS_LOCAL  int*)&lbT[bc * LB_STRIDE + br], 0, 0);
        }
        __builtin_amdgcn_s_wait_asynccnt(0);
#else
#pragma unroll
        for (int i = 0; i < 8; i++) {
            int e = tid + i * 256;
            int br = e >> 6, bc = e & 63;
            lbT[bc * LB_STRIDE + br] = Bm[(size_t)(kc + br) * HH + nbase + bc];
        }
#endif
        if (kc + 32 < K)  // gfx1250 global_prefetch for upcoming A stream
            __builtin_prefetch(Arow + kc + 32, 0, 1);
        __syncthreads();
#pragma unroll
        for (int ks = 0; ks < 8; ks++) {
            int k0 = kc + ks * 4;
            v2f a = *(const v2f*)&Arow[k0 + khalf];
#pragma unroll
            for (int t = 0; t < 4; t++) {
                v2f b = *(const v2f*)&lbT[(t * 16 + lrow) * LB_STRIDE + ks * 4 + khalf];
                acc[t] = __builtin_amdgcn_wmma_f32_16x16x4_f32(
                    false, a, false, b, (short)0, acc[t], false, false);
            }
        }
        __syncthreads();
    }
#pragma unroll
    for (int t = 0; t < 4; t++) {
#pragma unroll
        for (int v = 0; v < 8; v++) {
            int row = mbase + v + ((lane < 16) ? 0 : 8);
            int col = nbase + t * 16 + lrow;
            size_t idx = (size_t)row * HH + col;
            float val = acc[t][v];
            if (acc_flag) val += C[idx];
            if (bias)     val += bias[col];
            C[idx] = val;
        }
    }
}

// ---------------------------------------------------------------------------
// summary: s[j] = sigmoid(mean_n pos[n,j])
__global__ void k_summary(const float* __restrict__ pos, float* __restrict__ s) {
    int j = blockIdx.x * 256 + threadIdx.x;
    if (j >= HH) return;
    float acc = 0.f;
    for (int n = 0; n < NN; n++) acc += pos[(size_t)n * HH + j];
    acc *= (1.0f / (float)NN);
    s[j] = 1.0f / (1.0f + __expf(-acc));
}

// ws = W @ s
__global__ void k_wsvec(const float* __restrict__ W, const float* __restrict__ s,
                        float* __restrict__ wsv) {
    int i = blockIdx.x * 256 + threadIdx.x;
    if (i >= HH) return;
    float acc = 0.f;
    for (int j = 0; j < HH; j++) acc += W[(size_t)i * HH + j] * s[j];
    wsv[i] = acc;
}

__device__ __forceinline__ float logsig(float x) {
    return fminf(x, 0.0f) - log1pf(__expf(-fabsf(x)));
}

// accumulate sum_n logsig(pos_n . ws) and sum_n logsig(-(neg_n . ws))
__global__ void k_dgi(const float* __restrict__ pos, const float* __restrict__ neg,
                      const float* __restrict__ wsv, float* accPos, float* accNeg) {
    int row  = (blockIdx.x * 256 + threadIdx.x) >> 5;
    int lane = threadIdx.x & 31;
    if (row >= NN) return;
    float dp = 0.f, dn = 0.f;
    const float* pr = pos + (size_t)row * HH;
    const float* nr = neg + (size_t)row * HH;
    for (int j = lane; j < HH; j += 32) {
        float w = wsv[j];
        dp += pr[j] * w;
        dn += nr[j] * w;
    }
    for (int off = 16; off >= 1; off >>= 1) {
        dp += __shfl_xor(dp, off);
        dn += __shfl_xor(dn, off);
    }
    if (lane == 0) {
        atomicAdd(accPos, logsig(dp));
        atomicAdd(accNeg, logsig(-dn));
    }
}

__global__ void k_dgi_final(const float* __restrict__ accs, float* __restrict__ out) {
    if (threadIdx.x == 0) {
        float inv = 1.0f / (float)NN;
        float l1 = -(accs[0] + accs[1]) * inv;
        float l2 = -(accs[2] + accs[3]) * inv;
        out[2] = 0.5f * (l1 + l2);
        out[3] = 0.0f;
    }
}

// ---------------------------------------------------------------------------
// z[row] = emb_row / (||emb_row|| + 1e-8); rows 0..B-1 from a, B..2B-1 from b
__global__ void k_norm(const float* __restrict__ a, const float* __restrict__ b,
                       float* __restrict__ z) {
    int row  = (blockIdx.x * 256 + threadIdx.x) >> 5;
    int lane = threadIdx.x & 31;
    if (row >= 2 * BB) return;
    const float* srcp = (row < BB) ? (a + (size_t)row * HH) : (b + (size_t)(row - BB) * HH);
    float ss = 0.f;
    for (int j = lane; j < HH; j += 32) { float v = srcp[j]; ss += v * v; }
    for (int off = 16; off >= 1; off >>= 1) ss += __shfl_xor(ss, off);
    float scale = 1.0f / (sqrtf(ss) + 1e-8f);
    for (int j = lane; j < HH; j += 32) z[(size_t)row * HH + j] = srcp[j] * scale;
}

// Fused sim = z z^T tile + masked exp row-accumulation (MS loss core).
// Block handles 16 i-rows (A panel in LDS); 8 waves split the 256 j-tiles.
__global__ __launch_bounds__(256) void k_ms_sim(
    const float* __restrict__ z, const int* __restrict__ cid,
    float* __restrict__ psum, float* __restrict__ nsum) {
    __shared__ float lA[16 * HH];   // 48 KB
    const int tid = threadIdx.x, wave = tid >> 5, lane = tid & 31;
    const int lrow  = lane & 15;
    const int khalf = (lane < 16) ? 0 : 2;
    const int ibase = blockIdx.x * 16;

    for (int e = tid; e < 16 * HH; e += 256) lA[e] = z[(size_t)ibase * HH + e];
    __syncthreads();

    float pacc[8], nacc[8];
#pragma unroll
    for (int v = 0; v < 8; v++) { pacc[v] = 0.f; nacc[v] = 0.f; }
    const int myrow0 = ibase + ((lane < 16) ? 0 : 8);
    int labi[8];
#pragma unroll
    for (int v = 0; v < 8; v++) labi[v] = cid[(myrow0 + v) & (BB - 1)];

    const v8f vzero = {0.f,0.f,0.f,0.f,0.f,0.f,0.f,0.f};
    const float* lAr = &lA[lrow * HH];
    for (int jt = wave; jt < (2 * BB / 16); jt += 8) {
        int jbase = jt * 16;
        v8f acc = vzero;
        const float* Brow = z + (size_t)(jbase + lrow) * HH;  // lane's j row (B is z^T)
        for (int k0 = 0; k0 < HH; k0 += 4) {
            v2f a = *(const v2f*)&lAr[k0 + khalf];
            v2f b = *(const v2f*)&Brow[k0 + khalf];
            acc = __builtin_amdgcn_wmma_f32_16x16x4_f32(
                false, a, false, b, (short)0, acc, false, false);
        }
        int jg   = jbase + lrow;
        int labj = cid[jg & (BB - 1)];
#pragma unroll
        for (int v = 0; v < 8; v++) {
            int gi = myrow0 + v;
            float sim = acc[v];
            if (labi[v] == labj) {
                if (gi != jg) pacc[v] += __expf(-2.0f * (sim - 0.5f));
            } else {
                nacc[v] += __expf(50.0f * (sim - 0.5f));
            }
        }
    }
#pragma unroll
    for (int v = 0; v < 8; v++) {
        float p = pacc[v], n = nacc[v];
        for (int off = 8; off >= 1; off >>= 1) {
            p += __shfl_xor(p, off, 16);
            n += __shfl_xor(n, off, 16);
        }
        if (lrow == 0) {
            atomicAdd(&psum[myrow0 + v], p);
            atomicAdd(&nsum[myrow0 + v], n);
        }
    }
}

__global__ void k_ms_final(const float* __restrict__ psum, const float* __restrict__ nsum,
                           float* __restrict__ out) {
    __shared__ float st[256], sc[256];
    float tot = 0.f, cntv = 0.f;
    for (int i = threadIdx.x; i < 2 * BB; i += 256) {
        float pt  = log1pf(psum[i]) * 0.5f;    // / alpha (2)
        float nt  = log1pf(nsum[i]) * 0.02f;   // / beta (50)
        float per = pt + nt;
        tot  += per;
        cntv += (per > 0.0f) ? 1.0f : 0.0f;
    }
    st[threadIdx.x] = tot; sc[threadIdx.x] = cntv;
    __syncthreads();
    for (int off = 128; off >= 1; off >>= 1) {
        if (threadIdx.x < off) {
            st[threadIdx.x] += st[threadIdx.x + off];
            sc[threadIdx.x] += sc[threadIdx.x + off];
        }
        __syncthreads();
    }
    if (threadIdx.x == 0) out[0] = st[0] / fmaxf(sc[0], 1.0f);
}

// ---------------------------------------------------------------------------
extern "C" void kernel_launch(void* const* d_in, const int* in_sizes, int n_in,
                              void* d_out, int out_size, void* d_ws, size_t ws_size,
                              hipStream_t stream) {
    const float* te1   = (const float*)d_in[0];
    const float* te2   = (const float*)d_in[1];
    const float* basis = (const float*)d_in[2];
    const float* comp  = (const float*)d_in[3];
    const float* root  = (const float*)d_in[4];
    const float* bias  = (const float*)d_in[5];
    const float* dgiw  = (const float*)d_in[6];
    const int*   eidx  = (const int*)d_in[7];
    const int*   src   = eidx;
    const int*   dst   = eidx + EE;
    const int*   rel   = (const int*)d_in[8];
    const int*   cid   = (const int*)d_in[9];
    const int*   perm  = (const int*)d_in[10];
    float* out = (float*)d_out;

    float* ws = (float*)d_ws;
    size_t off = 0;
    float* wrel = ws + off; off += (size_t)RR * HH * HH;
    float* sums = ws + off; off += (size_t)NN * RR * HH;
    float* pos1 = ws + off; off += (size_t)NN * HH;
    float* neg1 = ws + off; off += (size_t)NN * HH;
    float* pos2 = ws + off; off += (size_t)NN * HH;
    float* neg2 = ws + off; off += (size_t)NN * HH;
    float* z    = ws + off; off += (size_t)2 * BB * HH;
    float* psum = ws + off; off += 2 * BB;
    float* nsum = ws + off; off += 2 * BB;
    float* svec = ws + off; off += HH;
    float* wsv  = ws + off; off += HH;
    float* dacc = ws + off; off += 8;
    unsigned* cnt = (unsigned*)(ws + off); off += NN * RR;

    // Basis combination + per-segment counts (graph-invariant across encodes)
    k_wrel<<<(RR * HH * HH) / 256, 256, 0, stream>>>(comp, basis, wrel);
    (void)hipMemsetAsync(cnt, 0, (size_t)NN * RR * sizeof(unsigned), stream);
    k_count<<<(EE + 255) / 256, 256, 0, stream>>>(dst, rel, cnt);
    (void)hipMemsetAsync(dacc, 0, 8 * sizeof(float), stream);

    const float* xs[4]   = { te1, te1, te2, te2 };
    const int*   pms[4]  = { nullptr, perm, nullptr, perm };
    float*       outs[4] = { pos1, neg1, pos2, neg2 };
    dim3 g1(NN / 128, HH / 64);
    for (int ei = 0; ei < 4; ei++) {
        (void)hipMemsetAsync(sums, 0, (size_t)NN * RR * HH * sizeof(float), stream);
        k_scatter<<<EE, 256, 0, stream>>>(xs[ei], src, dst, rel, pms[ei], sums);
        k_mean<<<(NN * RR * HH) / 256, 256, 0, stream>>>(sums, cnt);
        k_gemm<<<g1, 256, 0, stream>>>(sums, wrel, outs[ei], nullptr, nullptr, KREL, 0);
        k_gemm<<<g1, 256, 0, stream>>>(xs[ei], root, outs[ei], bias, pms[ei], HH, 1);
    }

    // DGI
    k_summary<<<3, 256, 0, stream>>>(pos1, svec);
    k_wsvec<<<3, 256, 0, stream>>>(dgiw, svec, wsv);
    k_dgi<<<NN / 8, 256, 0, stream>>>(pos1, neg1, wsv, &dacc[0], &dacc[1]);
    k_summary<<<3, 256, 0, stream>>>(pos2, svec);
    k_wsvec<<<3, 256, 0, stream>>>(dgiw, svec, wsv);
    k_dgi<<<NN / 8, 256, 0, stream>>>(pos2, neg2, wsv, &dacc[2], &dacc[3]);
    k_dgi_final<<<1, 64, 0, stream>>>(dacc, out);

    // text MS loss
    k_norm<<<(2 * BB) / 8, 256, 0, stream>>>(te1, te2, z);
    (void)hipMemsetAsync(psum, 0, 2 * BB * sizeof(float), stream);
    (void)hipMemsetAsync(nsum, 0, 2 * BB * sizeof(float), stream);
    k_ms_sim<<<2 * BB / 16, 256, 0, stream>>>(z, cid, psum, nsum);
    k_ms_final<<<1, 256, 0, stream>>>(psum, nsum, &out[0]);

    // graph MS loss
    k_norm<<<(2 * BB) / 8, 256, 0, stream>>>(pos1, pos2, z);
    (void)hipMemsetAsync(psum, 0, 2 * BB * sizeof(float), stream);
    (void)hipMemsetAsync(nsum, 0, 2 * BB * sizeof(float), stream);
    k_ms_sim<<<2 * BB / 16, 256, 0, stream>>>(z, cid, psum, nsum);
    k_ms_final<<<1, 256, 0, stream>>>(psum, nsum, &out[1]);
}